// CompositeLoss_10642928959652
// MI455X (gfx1250) — compile-verified
//
#include <hip/hip_runtime.h>
#include <stdint.h>

// ---------------------------------------------------------------------------
// CompositeLoss (weighted BCE + Dice) for pred(32,1,512,512) f32,
// target(32,1,512,512) i32 binary. Output: [total, bce, dice] f32.
//
// MI455X reasoning: 64 MB compulsory traffic @ 23.3 TB/s ~ 2.8 us floor.
// - target rows (+/-1 halo, clamp-replicated == clipped-window min/max)
//   staged to LDS via global_load_async_to_lds_b128 (ASYNCcnt path)
// - pred streamed with coalesced float4 loads
// - 3x3 min/max done separably (6 vertical columns -> 4 horizontal windows)
// - wave partial-sum fold via V_WMMA_F32_16X16X4_F32 (ones * B: 32->16 lanes)
// - deterministic: block partials -> fixed-order f64 tree in a 1-block kernel
// ---------------------------------------------------------------------------

typedef __attribute__((ext_vector_type(2))) float v2f;
typedef __attribute__((ext_vector_type(8))) float v8f;

#define IMG_B  32
#define H      512
#define W      512
#define ROWS   8
#define BLK    256
#define NBLOCKS (IMG_B * (H / ROWS))          /* 2048 */
#define NPIX   (8388608.0)                    /* 32*512*512 */

__global__ __launch_bounds__(BLK) void composite_loss_partial(
    const float* __restrict__ pred, const int* __restrict__ tgt,
    float* __restrict__ partial) {
  __shared__ __attribute__((aligned(16))) int t_lds[(ROWS + 2) * W]; // 20 KB
  __shared__ float red[32];

  const int bid   = blockIdx.x;
  const int img   = bid >> 6;       // 64 strips per image
  const int strip = bid & 63;
  const int r0    = strip * ROWS;
  const int tid   = threadIdx.x;

  const long long plane = (long long)H * W;
  const int*   tbase = tgt  + (long long)img * plane;
  const float* pbase = pred + (long long)img * plane;

  // ---- async stage: rows clamp(r0-1 .. r0+8) -> LDS (10*512 ints, b128) ----
  #pragma unroll
  for (int i = 0; i < 5; ++i) {
    int chunk = tid + i * BLK;          // 1280 chunks x 16B
    int lrow  = chunk >> 7;             // 0..9 (128 chunks per row)
    int cc    = (chunk & 127) << 2;     // starting int column
    int grow  = r0 - 1 + lrow;
    grow = grow < 0 ? 0 : (grow > H - 1 ? H - 1 : grow);
    unsigned lds_off = (unsigned)(uintptr_t)(&t_lds[lrow * W + cc]);
    const int* gp = tbase + (long long)grow * W + cc;
    asm volatile("global_load_async_to_lds_b128 %0, %1, off"
                 :: "v"(lds_off), "v"(gp)
                 : "memory");
  }
  asm volatile("s_wait_asynccnt 0" ::: "memory");
  __syncthreads();

  // ---- per-pixel: boundary-weighted BCE + dice sums (16 px/thread) ----
  float s_bce = 0.0f, s_i = 0.0f, s_p = 0.0f, s_t = 0.0f;
  #pragma unroll
  for (int i = 0; i < 4; ++i) {
    int chunk = tid + i * BLK;          // 1024 chunks of 4 px
    int lr    = chunk >> 7;             // local row 0..7
    int col0  = (chunk & 127) << 2;     // column of first pixel
    const float4 pv =
        *(const float4*)(pbase + (long long)(r0 + lr) * W + col0);

    int vmn[6], vmx[6], tc[6];
    #pragma unroll
    for (int j = 0; j < 6; ++j) {
      int x = col0 - 1 + j;
      x = x < 0 ? 0 : (x > W - 1 ? W - 1 : x);   // clamp == clipped window
      int a = t_lds[(lr + 0) * W + x];
      int b = t_lds[(lr + 1) * W + x];
      int c = t_lds[(lr + 2) * W + x];
      tc[j]  = b;                                  // center row cache
      vmn[j] = min(a, min(b, c));
      vmx[j] = max(a, max(b, c));
    }
    float pvals[4] = {pv.x, pv.y, pv.z, pv.w};
    #pragma unroll
    for (int j = 0; j < 4; ++j) {
      int t  = tc[j + 1];
      int mn = min(vmn[j], min(vmn[j + 1], vmn[j + 2]));
      int mx = max(vmx[j], max(vmx[j + 1], vmx[j + 2]));
      float w  = (mx != mn) ? 3.0f : 1.0f;
      float pr = pvals[j];
      float pc = fminf(fmaxf(pr, 1e-7f), 1.0f - 1e-7f);
      float arg = t ? pc : (1.0f - pc);
      s_bce = fmaf(w, -__logf(arg), s_bce);
      float tf = (float)t;
      s_i = fmaf(pr, tf, s_i);
      s_p += pr;
      s_t += tf;
    }
  }

  // ---- wave reduction: WMMA fold 32->16 lanes, then shfl ----
  v2f a_ones; a_ones.x = 1.0f; a_ones.y = 1.0f;   // A = ones(16x4)
  float sums[4] = {s_bce, s_i, s_p, s_t};
  float wres[4];
  #pragma unroll
  for (int k = 0; k < 4; ++k) {
    v2f b; b.x = sums[k]; b.y = 0.0f;             // B[0][n]=v[n], B[2][n]=v[n+16]
    v8f c = {};
    v8f d = __builtin_amdgcn_wmma_f32_16x16x4_f32(
        false, a_ones, false, b, (short)0, c, false, false);
    float v = d[0];                               // v[n] + v[n+16], dup per half
    v += __shfl_xor(v, 8, 32);
    v += __shfl_xor(v, 4, 32);
    v += __shfl_xor(v, 2, 32);
    v += __shfl_xor(v, 1, 32);
    wres[k] = v;                                  // wave total (all lanes)
  }
  int wid = tid >> 5, lane = tid & 31;
  if (lane == 0) {
    red[wid * 4 + 0] = wres[0];
    red[wid * 4 + 1] = wres[1];
    red[wid * 4 + 2] = wres[2];
    red[wid * 4 + 3] = wres[3];
  }
  __syncthreads();
  if (tid < 32) {                // 8 waves x 4 accumulators, stride-4 classes
    float v = red[tid];
    v += __shfl_xor(v, 16, 32);
    v += __shfl_xor(v, 8, 32);
    v += __shfl_xor(v, 4, 32);
    if (tid < 4) partial[bid * 4 + tid] = v;
  }
}

// ---- deterministic fixed-order final reduction (f64) + scalar epilogue ----
__global__ __launch_bounds__(BLK) void composite_loss_final(
    const float* __restrict__ partial, float* __restrict__ out, int nblocks) {
  __shared__ double lred[8][4];
  double s[4] = {0.0, 0.0, 0.0, 0.0};
  for (int i = threadIdx.x; i < nblocks; i += BLK) {
    s[0] += (double)partial[i * 4 + 0];
    s[1] += (double)partial[i * 4 + 1];
    s[2] += (double)partial[i * 4 + 2];
    s[3] += (double)partial[i * 4 + 3];
  }
  #pragma unroll
  for (int m = 16; m >= 1; m >>= 1) {
    s[0] += __shfl_xor(s[0], m, 32);
    s[1] += __shfl_xor(s[1], m, 32);
    s[2] += __shfl_xor(s[2], m, 32);
    s[3] += __shfl_xor(s[3], m, 32);
  }
  int wid = threadIdx.x >> 5, lane = threadIdx.x & 31;
  if (lane == 0) {
    lred[wid][0] = s[0]; lred[wid][1] = s[1];
    lred[wid][2] = s[2]; lred[wid][3] = s[3];
  }
  __syncthreads();
  if (threadIdx.x == 0) {
    double t0 = 0, t1 = 0, t2 = 0, t3 = 0;
    for (int w = 0; w < 8; ++w) {        // fixed order
      t0 += lred[w][0]; t1 += lred[w][1]; t2 += lred[w][2]; t3 += lred[w][3];
    }
    double bce  = t0 / NPIX;
    double dice = 1.0 - (2.0 * t1 + 1e-6) / (t2 + t3 + 1e-6);
    out[0] = (float)(0.5 * bce + 0.5 * dice);
    out[1] = (float)bce;
    out[2] = (float)dice;
  }
}

extern "C" void kernel_launch(void* const* d_in, const int* in_sizes, int n_in,
                              void* d_out, int out_size, void* d_ws, size_t ws_size,
                              hipStream_t stream) {
  (void)in_sizes; (void)n_in; (void)out_size; (void)ws_size;
  const float* pred = (const float*)d_in[0];
  const int*   tgt  = (const int*)d_in[1];
  float* partial = (float*)d_ws;                 // 2048 * 4 floats = 32 KB
  composite_loss_partial<<<NBLOCKS, BLK, 0, stream>>>(pred, tgt, partial);
  composite_loss_final<<<1, BLK, 0, stream>>>(partial, (float*)d_out, NBLOCKS);
}